// GradOptimLayerAttn2D_6665789243616
// MI455X (gfx1250) — compile-verified
//
#include <hip/hip_runtime.h>
#include <math.h>

// Problem constants (from reference)
#define B_TOT 2048
#define V_TOT 2048
#define C_TOT 512
#define M_CNT 8
#define H_CNT 16

// Tiling
#define TILE_B 16          // b rows per block (one WMMA N-tile)
#define NTHREADS 256       // 8 wave32 waves
#define NCCHUNK 4          // c-chunks across gridDim.y
#define CCHUNK (C_TOT / NCCHUNK)          // 128 c per block, 16 per wave
#define TAILCOLS ((V_TOT - C_TOT) / NCCHUNK) // 384 pass-through cols per block
#define LDS_STRIDE 2052    // 2048 + 4 floats: (idx + 4*row) % 64 -> conflict-free

typedef __attribute__((ext_vector_type(2))) float v2f;
typedef __attribute__((ext_vector_type(4))) float v4f;
typedef __attribute__((ext_vector_type(8))) float v8f;
typedef int vint4 __attribute__((vector_size(16)));   // matches builtin proto

#if __has_builtin(__builtin_amdgcn_tanhf)
#define FAST_TANH(x) __builtin_amdgcn_tanhf(x)
#else
#define FAST_TANH(x) tanhf(x)
#endif

#if __has_builtin(__builtin_amdgcn_global_load_async_to_lds_b128)
#define HAVE_ASYNC_LDS 1
#endif

__global__ void __launch_bounds__(NTHREADS)
attn_correct_kernel(const float* __restrict__ preds,
                    const int*   __restrict__ mask_ids,
                    const float* __restrict__ W1,
                    const float* __restrict__ b1,
                    const float* __restrict__ W2,
                    float*       __restrict__ out)
{
  extern __shared__ float smem[];            // TILE_B * LDS_STRIDE floats
  const int tid    = threadIdx.x;
  const int b0     = blockIdx.x * TILE_B;    // 128 b-tiles
  const int cchunk = blockIdx.y;             // 0..3

  // ---------------- Stage 16 preds rows (128 KB) into LDS ----------------
  // 8192 16-byte chunks, 32 per thread. Async on gfx1250 (no VGPR round-trip).
  #pragma unroll 4
  for (int i = 0; i < (TILE_B * V_TOT / 4) / NTHREADS; ++i) {
    const int q    = tid + i * NTHREADS;
    const int row  = q >> 9;        // /512 chunks per row
    const int col4 = q & 511;       // float4 index within row
    const float* gsrc = preds + (size_t)(b0 + row) * V_TOT + col4 * 4;
    float*       ldst = smem + row * LDS_STRIDE + col4 * 4;
#ifdef HAVE_ASYNC_LDS
    __builtin_amdgcn_global_load_async_to_lds_b128(
        (__attribute__((address_space(1))) vint4*)gsrc,
        (__attribute__((address_space(3))) vint4*)ldst, 0, 0);
#else
    *(v4f*)ldst = *(const v4f*)gsrc;
#endif
  }
#ifdef HAVE_ASYNC_LDS
#if __has_builtin(__builtin_amdgcn_s_wait_asynccnt)
  __builtin_amdgcn_s_wait_asynccnt(0);
#else
  asm volatile("s_wait_asynccnt 0" ::: "memory");
#endif
#endif
  __syncthreads();

  // ---------------- Per-wave: c is wave-uniform; lanes span b-tile ----------
  const int lane   = tid & 31;
  const int wave   = tid >> 5;     // 0..7
  const int lane16 = lane & 15;    // b index within tile (N axis)
  const int upper  = lane >> 4;    // D rows h = j + 8*upper
  const int hsel   = upper ? 8 : 0;
  const float* rowp = smem + lane16 * LDS_STRIDE;
  const int cbase  = cchunk * CCHUNK;

  for (int it = 0; it < CCHUNK / 8; ++it) {
    const int c = cbase + wave + it * 8;

    // A = W1[c]^T in f32 16x4 A-layout (M=h in lanes, K in VGPRs/halves).
    // Lanes 16-31 hold K=2,3 -> must be zero (K padding).
    float a0 = 0.f, a1 = 0.f;
    if (!upper) {
      a0 = W1[c * (2 * H_CNT) + lane16];          // W1[c,0,h=lane]
      a1 = W1[c * (2 * H_CNT) + H_CNT + lane16];  // W1[c,1,h=lane]
    }
    v2f A; A.x = a0; A.y = a1;

    // C accumulator = bias b1[c,h] broadcast over b; score weights W2[c,h].
    v8f Cacc;
    float wj[8];
    #pragma unroll
    for (int j = 0; j < 8; ++j) {
      Cacc[j] = b1[c * H_CNT + hsel + j];
      wj[j]   = W2[c * H_CNT + hsel + j];
    }

    const float h_a = rowp[c];
    float sm[M_CNT], hmv[M_CNT];

    #pragma unroll
    for (int m = 0; m < M_CNT; ++m) {
      const int   idx = mask_ids[c * M_CNT + m];  // in [512, 2048)
      const float hm  = rowp[idx];
      hmv[m] = hm;

#if __has_builtin(__builtin_amdgcn_wmma_f32_16x16x4_f32)
      // B = feat (K x N=b): row K=0 = h_a[b=lane16], K=1 = h_m[b=lane16].
      // Upper-lane rows K=2,3 are don't-care (A's K=2,3 columns are zero).
      v2f Bv; Bv.x = h_a; Bv.y = hm;
      // D[h, b] = W1[c]^T x feat + b1   (pre-activation, transposed)
      v8f d = __builtin_amdgcn_wmma_f32_16x16x4_f32(
          false, A, false, Bv, (short)0, Cacc, false, false);
#else
      // Exact per-lane emulation of the D layout above.
      v8f d;
      #pragma unroll
      for (int j = 0; j < 8; ++j)
        d[j] = fmaf(h_a, W1[c * (2 * H_CNT) + hsel + j],
               fmaf(hm,  W1[c * (2 * H_CNT) + H_CNT + hsel + j], Cacc[j]));
#endif

      // hid = tanh(pre); partial score over this lane's 8 h values
      float partial = 0.f;
      #pragma unroll
      for (int j = 0; j < 8; ++j)
        partial = fmaf(FAST_TANH(d[j]), wj[j], partial);
      // combine h=0..7 (lanes 0-15) with h=8..15 (lanes 16-31)
      partial += __shfl_xor(partial, 16, 32);
      sm[m] = partial;   // b2[c] omitted: uniform over m -> softmax-invariant
    }

    // softmax over m + attention-weighted h_m
    float mx = sm[0];
    #pragma unroll
    for (int m = 1; m < M_CNT; ++m) mx = fmaxf(mx, sm[m]);
    float esum = 0.f, wsum = 0.f;
    #pragma unroll
    for (int m = 0; m < M_CNT; ++m) {
      const float e = __expf(sm[m] - mx);
      esum += e;
      wsum  = fmaf(e, hmv[m], wsum);
    }
    const float corrected = fmaxf(h_a, wsum / esum + 1e-6f);
    if (!upper)
      out[(size_t)(b0 + lane16) * V_TOT + c] = corrected;
  }

  // ---------------- Pass-through columns [512, 2048) -----------------------
  const int colbase = C_TOT + cchunk * TAILCOLS;
  for (int i = tid; i < TILE_B * TAILCOLS; i += NTHREADS) {
    const int r   = i / TAILCOLS;
    const int col = colbase + (i - r * TAILCOLS);
    out[(size_t)(b0 + r) * V_TOT + col] = smem[r * LDS_STRIDE + col];
  }
}

extern "C" void kernel_launch(void* const* d_in, const int* in_sizes, int n_in,
                              void* d_out, int out_size, void* d_ws, size_t ws_size,
                              hipStream_t stream) {
  (void)in_sizes; (void)n_in; (void)d_ws; (void)ws_size; (void)out_size;
  const float* preds    = (const float*)d_in[0];
  // d_in[1] = ground_truth (unused by reference)
  const int*   mask_ids = (const int*)  d_in[2];
  const float* W1       = (const float*)d_in[3];
  const float* b1       = (const float*)d_in[4];
  const float* W2       = (const float*)d_in[5];
  // d_in[6] = b2 : uniform over m -> cancels in softmax, unused
  float* out = (float*)d_out;

  dim3 grid(B_TOT / TILE_B, NCCHUNK);   // 128 x 4 = 512 blocks (4096 waves)
  dim3 block(NTHREADS);
  size_t shmem = (size_t)TILE_B * LDS_STRIDE * sizeof(float);  // 131,328 B
  hipLaunchKernelGGL(attn_correct_kernel, grid, block, shmem, stream,
                     preds, mask_ids, W1, b1, W2, out);
}